// HungarianMatcher_78245714198741
// MI455X (gfx1250) — compile-verified
//
#include <hip/hip_runtime.h>
#include <hip/hip_bf16.h>

// ---------------------------------------------------------------------------
// DETR HungarianMatcher cost matrix for MI455X (gfx1250, wave32).
//
//   out[i,j] = L1(box_i, tbox_j) + focal_class_cost(sigmoid(logit[i, id_j]))
//              - GIoU(xyxy_i, xyxy_j)
//
// Strategy:
//  * Precompute CC[i][c] (focal class cost, 16000x92 padded) -> kills 22x of
//    the transcendental work (1.46M instead of 32M sigmoid/log chains).
//  * cost_class tile = CC x OneHot(ids)  via V_WMMA_F32_16X16X4_F32 (fp32,
//    bit-exact: one-hot products are x*1 or x*0). Runs on matrix pipe,
//    co-executes with GIoU VALU math.
//  * Precompute per-row / per-target box records (cxcywh, xyxy, area).
//  * Main kernel: 1 wave = 16-row tile, A fragments register-resident
//    (46 VGPRs), sweeps target tiles; 128MB of output stores is the
//    bandwidth floor (~5.5us @ 23.3 TB/s).
// ---------------------------------------------------------------------------

typedef float v2f __attribute__((ext_vector_type(2)));
typedef float v8f __attribute__((ext_vector_type(8)));

#define CPAD    92          // 91 classes padded to multiple of 4
#define KSTEPS  23          // 92 / 4
#define ALPHA_F 0.25f
#define EPS_F   1e-8f

// ---------------------------------------------------------------------------
// Precompute focal class-cost table CC[R][CPAD] (pad columns = 0).
// CC = ALPHA*(1-p)^2*(-log(p+eps)) - (1-ALPHA)*p^2*(-log(1-p+eps))
// ---------------------------------------------------------------------------
__global__ void hm_precompute_cc(const float* __restrict__ logits,
                                 float* __restrict__ cc, int R, int C) {
    int idx = blockIdx.x * 256 + threadIdx.x;
    if (idx >= R * CPAD) return;
    int i = idx / CPAD, c = idx - i * CPAD;
    float v = 0.0f;
    if (c < C) {
        float x = logits[(size_t)i * C + c];
        float p = __builtin_amdgcn_rcpf(1.0f + __expf(-x));   // sigmoid
        float om = 1.0f - p;
        float pos = ALPHA_F * om * om * (-__logf(p + EPS_F));
        float neg = (1.0f - ALPHA_F) * p * p * (-__logf(om + EPS_F));
        v = pos - neg;
    }
    cc[idx] = v;
}

// ---------------------------------------------------------------------------
// Precompute box records, stride 12 floats:
//   [cx, cy, w, h, x0, y0, x1, y1, area, cls_bits, 0, 0]
// ---------------------------------------------------------------------------
__global__ void hm_precompute_boxes(const float* __restrict__ boxes,
                                    const int* __restrict__ ids,
                                    float* __restrict__ rec, int N, int hasIds) {
    int i = blockIdx.x * 256 + threadIdx.x;
    if (i >= N) return;
    float cx = boxes[(size_t)i * 4 + 0];
    float cy = boxes[(size_t)i * 4 + 1];
    float w  = boxes[(size_t)i * 4 + 2];
    float h  = boxes[(size_t)i * 4 + 3];
    float x0 = cx - 0.5f * w, y0 = cy - 0.5f * h;
    float x1 = cx + 0.5f * w, y1 = cy + 0.5f * h;
    float area = (x1 - x0) * (y1 - y0);           // matches reference exactly
    float4* r = (float4*)(rec + (size_t)i * 12);
    r[0] = make_float4(cx, cy, w, h);
    r[1] = make_float4(x0, y0, x1, y1);
    float clsb = hasIds ? __int_as_float(ids[i]) : 0.0f;
    r[2] = make_float4(area, clsb, 0.0f, 0.0f);
}

// ---------------------------------------------------------------------------
// Main fused kernel: WMMA class cost + VALU bbox/giou.
// Block = 128 threads = 4 waves; each wave owns one (rowTile, jChunk) task.
// ---------------------------------------------------------------------------
__global__ __launch_bounds__(128)
void hm_main(const float* __restrict__ cc,     // [R][CPAD]
             const float* __restrict__ rbox,   // [R][12]
             const float* __restrict__ tbox,   // [T][12]
             float* __restrict__ out,          // [R][T]
             int R, int T, int jChunks, int jtPerChunk) {
    const int lane = threadIdx.x & 31;
    const int wave = threadIdx.x >> 5;
    const int task = blockIdx.x * 4 + wave;
    const int rowTile = task / jChunks;
    const int jChunk  = task - rowTile * jChunks;
    if (rowTile * 16 >= R) return;               // uniform per-wave exit

    const int nJT = T >> 4;
    const int jt0 = jChunk * jtPerChunk;
    const int jt1 = (jt0 + jtPerChunk < nJT) ? (jt0 + jtPerChunk) : nJT;

    const int h  = lane >> 4;                    // half-wave id (0/1)
    const int ml = lane & 15;                    // M index for A / N index for B

    // ---- stage this wave's 16 row records into LDS (broadcast reads later)
    __shared__ __align__(16) float lrow[4][16 * 12];
    if (lane < 16) {
        const float4* src = (const float4*)(rbox + (size_t)(rowTile * 16 + lane) * 12);
        float4 a0 = src[0], a1 = src[1], a2 = src[2];
        float4* dst = (float4*)(&lrow[wave][lane * 12]);
        dst[0] = a0; dst[1] = a1; dst[2] = a2;
    }
    asm volatile("s_wait_dscnt 0" ::: "memory"); // same-wave LDS WAR/RAW safety

    // ---- preload A fragments (CC slab, 16 x 92) into 46 VGPRs
    // ISA A-layout 16x4 f32: lanes 0-15 hold K=0,1; lanes 16-31 hold K=2,3.
    v2f areg[KSTEPS];
    const float* ccp = cc + (size_t)(rowTile * 16 + ml) * CPAD + h * 2;
    #pragma unroll
    for (int s = 0; s < KSTEPS; ++s)
        areg[s] = *(const v2f*)(ccp + s * 4);

    float* lr = &lrow[wave][0];

    for (int jt = jt0; jt < jt1; ++jt) {
        const int j = jt * 16 + ml;              // this lane's target (N index)
        const float4* tp = (const float4*)(tbox + (size_t)j * 12);
        const float4 t0 = tp[0];                 // cx cy w h
        const float4 t1 = tp[1];                 // x0 y0 x1 y1
        const float4 t2 = tp[2];                 // area, cls_bits
        const int cls = __float_as_int(t2.y);

        // ---- one-hot B build + 23 chained fp32 WMMAs (class cost tile)
        // B-layout 4x16 f32: VGPR0 = rows K=0 (lanes 0-15) / K=1 (16-31),
        //                    VGPR1 = rows K=2 / K=3.
        v8f acc = {0.f, 0.f, 0.f, 0.f, 0.f, 0.f, 0.f, 0.f};
        #pragma unroll
        for (int s = 0; s < KSTEPS; ++s) {
            v2f b;
            b.x = (cls == (4 * s + h))     ? 1.0f : 0.0f;
            b.y = (cls == (4 * s + 2 + h)) ? 1.0f : 0.0f;
            acc = __builtin_amdgcn_wmma_f32_16x16x4_f32(
                false, areg[s], false, b, (short)0, acc, false, false);
        }

        // ---- elementwise bbox-L1 + GIoU directly in WMMA C layout:
        // slot v: lanes 0-15 -> M=v, lanes 16-31 -> M=v+8; N = lane&15.
        #pragma unroll
        for (int v = 0; v < 8; ++v) {
            const int mlocal = v + 8 * h;
            const float4 r0 = *(const float4*)(lr + mlocal * 12);     // cxcywh
            const float4 r1 = *(const float4*)(lr + mlocal * 12 + 4); // xyxy
            const float rarea = lr[mlocal * 12 + 8];

            float cbbox = fabsf(r0.x - t0.x) + fabsf(r0.y - t0.y)
                        + fabsf(r0.z - t0.z) + fabsf(r0.w - t0.w);

            float ltx = fmaxf(r1.x, t1.x), lty = fmaxf(r1.y, t1.y);
            float rbx = fminf(r1.z, t1.z), rby = fminf(r1.w, t1.w);
            float iw = fmaxf(rbx - ltx, 0.0f), ih = fmaxf(rby - lty, 0.0f);
            float inter = iw * ih;
            float uni = rarea + t2.x - inter;

            float lex = fminf(r1.x, t1.x), ley = fminf(r1.y, t1.y);
            float rex = fmaxf(r1.z, t1.z), rey = fmaxf(r1.w, t1.w);
            float ew = fmaxf(rex - lex, 0.0f), eh = fmaxf(rey - ley, 0.0f);
            float earea = ew * eh;

            float giou = inter * __builtin_amdgcn_rcpf(uni)
                       - (earea - uni) * __builtin_amdgcn_rcpf(earea);

            const int m = rowTile * 16 + mlocal;
            out[(size_t)m * T + j] = cbbox + acc[v] - giou;
        }
    }
}

// ---------------------------------------------------------------------------
// Fallback: fully inline, one thread per output element.
// ---------------------------------------------------------------------------
__global__ void hm_naive(const float* __restrict__ logits,
                         const float* __restrict__ boxes,
                         const int* __restrict__ ids,
                         const float* __restrict__ tbb,
                         float* __restrict__ out,
                         int R, int C, long long T) {
    long long idx = (long long)blockIdx.x * 256 + threadIdx.x;
    if (idx >= (long long)R * T) return;
    int i = (int)(idx / T), j = (int)(idx - (long long)i * T);

    int cls = ids[j];
    float x = logits[(size_t)i * C + cls];
    float p = __builtin_amdgcn_rcpf(1.0f + __expf(-x));
    float om = 1.0f - p;
    float cclass = ALPHA_F * om * om * (-__logf(p + EPS_F))
                 - (1.0f - ALPHA_F) * p * p * (-__logf(om + EPS_F));

    float acx = boxes[(size_t)i * 4 + 0], acy = boxes[(size_t)i * 4 + 1];
    float aw  = boxes[(size_t)i * 4 + 2], ah  = boxes[(size_t)i * 4 + 3];
    float bcx = tbb[(size_t)j * 4 + 0],  bcy = tbb[(size_t)j * 4 + 1];
    float bw  = tbb[(size_t)j * 4 + 2],  bh  = tbb[(size_t)j * 4 + 3];

    float cbbox = fabsf(acx - bcx) + fabsf(acy - bcy)
                + fabsf(aw - bw) + fabsf(ah - bh);

    float ax0 = acx - 0.5f * aw, ay0 = acy - 0.5f * ah;
    float ax1 = acx + 0.5f * aw, ay1 = acy + 0.5f * ah;
    float bx0 = bcx - 0.5f * bw, by0 = bcy - 0.5f * bh;
    float bx1 = bcx + 0.5f * bw, by1 = bcy + 0.5f * bh;
    float areaA = (ax1 - ax0) * (ay1 - ay0);
    float areaB = (bx1 - bx0) * (by1 - by0);

    float iw = fmaxf(fminf(ax1, bx1) - fmaxf(ax0, bx0), 0.0f);
    float ih = fmaxf(fminf(ay1, by1) - fmaxf(ay0, by0), 0.0f);
    float inter = iw * ih;
    float uni = areaA + areaB - inter;
    float ew = fmaxf(fmaxf(ax1, bx1) - fminf(ax0, bx0), 0.0f);
    float eh = fmaxf(fmaxf(ay1, by1) - fminf(ay0, by0), 0.0f);
    float earea = ew * eh;
    float giou = inter * __builtin_amdgcn_rcpf(uni)
               - (earea - uni) * __builtin_amdgcn_rcpf(earea);

    out[idx] = cbbox + cclass - giou;
}

// ---------------------------------------------------------------------------
extern "C" void kernel_launch(void* const* d_in, const int* in_sizes, int n_in,
                              void* d_out, int out_size, void* d_ws, size_t ws_size,
                              hipStream_t stream) {
    const float* logits = (const float*)d_in[0];   // (bs,Q,C) fp32
    const float* boxes  = (const float*)d_in[1];   // (bs,Q,4) fp32
    const int*   ids    = (const int*)d_in[2];     // (T,) int32
    const float* tbb    = (const float*)d_in[3];   // (T,4) fp32
    float* out = (float*)d_out;

    const int R = in_sizes[1] / 4;                 // bs*Q = 16000
    const int T = in_sizes[3] / 4;                 // 2000
    const int C = in_sizes[0] / R;                 // 91

    const size_t ccBytes = (size_t)R * CPAD * sizeof(float);   // 5.888 MB
    const size_t rBytes  = (size_t)R * 12 * sizeof(float);     // 768 KB
    const size_t tBytes  = (size_t)T * 12 * sizeof(float);     // 96 KB
    const size_t need = ccBytes + rBytes + tBytes;

    const bool fast = (ws_size >= need) && (R % 16 == 0) && (T % 16 == 0)
                   && (C <= CPAD) && (C > CPAD - 4);

    if (fast) {
        float* cc   = (float*)d_ws;
        float* rrec = (float*)((char*)d_ws + ccBytes);
        float* trec = (float*)((char*)d_ws + ccBytes + rBytes);

        hm_precompute_cc<<<(R * CPAD + 255) / 256, 256, 0, stream>>>(logits, cc, R, C);
        hm_precompute_boxes<<<(R + 255) / 256, 256, 0, stream>>>(boxes, nullptr, rrec, R, 0);
        hm_precompute_boxes<<<(T + 255) / 256, 256, 0, stream>>>(tbb, ids, trec, T, 1);

        const int nJT = T / 16;                          // 125
        const int jtPerChunk = (nJT % 5 == 0) ? nJT / 5 : nJT;  // 25
        const int jChunks = (nJT + jtPerChunk - 1) / jtPerChunk; // 5
        const int tasks = (R / 16) * jChunks;            // 5000 waves
        const int blocks = (tasks + 3) / 4;              // 4 waves/block
        hm_main<<<blocks, 128, 0, stream>>>(cc, rrec, trec, out, R, T, jChunks, jtPerChunk);
    } else {
        long long total = (long long)R * T;
        hm_naive<<<(unsigned)((total + 255) / 256), 256, 0, stream>>>(
            logits, boxes, ids, tbb, out, R, C, (long long)T);
    }
}